// CostVolume_86131274154867
// MI455X (gfx1250) — compile-verified
//
#include <hip/hip_runtime.h>
#include <math.h>

// Problem constants (from setup_inputs): N=2, C=256, H=55, W=128, rx=ry=2.
#define NB   2
#define CHN  256
#define HH   55
#define WW   128
#define HWSZ (HH * WW)        // 7040
#define NHW  (NB * HWSZ)      // 14080 pixels
#define CVN  (NHW * 25)       // 352000 cv elements
#define OUT_WTA (CVN)         // wta_val offset
#define OUT_XY  (CVN + NHW)   // wta_xy offset ([2,N,H,W], written as float)

typedef __attribute__((ext_vector_type(2))) float v2f;
typedef __attribute__((ext_vector_type(4))) float v4f;
typedef __attribute__((ext_vector_type(8))) float v8f;

// ---------------------------------------------------------------------------
// Kernel 1: transpose [N,C,H,W] -> [N,H,W,C] so per-pixel channel vectors are
// contiguous 1KB blocks (coalesced b128/b64 streams for the WMMA B operand).
// 32x32 tile through LDS, 32x8 threads, 4 rows per thread.
// ---------------------------------------------------------------------------
__global__ void xpose_kernel(const float* __restrict__ src, float* __restrict__ dst) {
    __shared__ float tile[32][33];
    const int plane = blockIdx.y;              // n*H + h
    const int n = plane / HH, h = plane % HH;
    const int t  = blockIdx.x;                 // 8 c-tiles x 4 w-tiles
    const int c0 = (t >> 2) * 32, w0 = (t & 3) * 32;
    const int tx = threadIdx.x, ty = threadIdx.y;
    for (int r = 0; r < 32; r += 8)
        tile[ty + r][tx] = src[((long)(n * CHN + c0 + ty + r) * HH + h) * WW + w0 + tx];
    __syncthreads();
    for (int r = 0; r < 32; r += 8)
        dst[((long)plane * WW + (w0 + ty + r)) * CHN + c0 + tx] = tile[tx][ty + r];
}

// ---------------------------------------------------------------------------
// Kernel 2: one wave (32 threads, single-wave workgroup -> free barriers) per
// pixel. Computes the 6x6 patch of channel-dots D with V_WMMA_F32_16X16X4_F32
// (exact fp32): N-dim = 16 patch positions (3 accumulators cover 36), A = f1
// chunk broadcast across rows, K-loop over 64 chunks of 4 channels. Then the
// bilinear 4-tap combine per window offset and the 25-way WTA reduction.
// ---------------------------------------------------------------------------
__global__ __launch_bounds__(32) void cv_kernel(const float* __restrict__ flow,
                                                const float* __restrict__ f1t,
                                                const float* __restrict__ f2t,
                                                float* __restrict__ out) {
    __shared__ float sF1[CHN];
    __shared__ float sD[40];

    const int pix  = blockIdx.x;
    const int n    = pix / HWSZ;
    const int rem  = pix % HWSZ;
    const int h    = rem / WW, w = rem % WW;
    const int lane = threadIdx.x;

    // Stage this pixel's f1 channel vector (contiguous after transpose) in LDS.
    {
        const float* p = f1t + (long)pix * CHN + lane * 8;
        v4f a = *(const v4f*)p;
        v4f b = *(const v4f*)(p + 4);
        *(v4f*)(sF1 + lane * 8)     = a;
        *(v4f*)(sF1 + lane * 8 + 4) = b;
    }
    __syncthreads();

    const float fx  = flow[((long)(n * 2 + 0) * HH + h) * WW + w];
    const float fy  = flow[((long)(n * 2 + 1) * HH + h) * WW + w];
    const float xf  = (float)w + fx, yf = (float)h + fy;
    const float x0f = floorf(xf),    y0f = floorf(yf);
    const int   x0  = (int)x0f,      y0  = (int)y0f;
    const float wx  = xf - x0f,      wy  = yf - y0f;

    // WMMA operand layout (wave32): A 16x4 f32: lanes 0-15 hold K=0,1 and
    // lanes 16-31 hold K=2,3 (2 VGPRs). B assumed mirrored: lane half selects
    // K pair, lane%16 selects column N (= patch position).
    const int nl    = lane & 15;
    const int khalf = (lane >> 4) * 2;

    long  base[3];
    float mask[3];
#pragma unroll
    for (int g = 0; g < 3; ++g) {
        const int p  = g * 16 + nl;          // patch position j*6+i
        const int j  = p / 6, i = p - j * 6;
        const int yy = y0 - 2 + j, xx = x0 - 2 + i;
        const bool ok = (p < 36) & (xx >= 0) & (xx < WW) & (yy >= 0) & (yy < HH);
        const int yc = min(max(yy, 0), HH - 1);
        const int xc = min(max(xx, 0), WW - 1);
        base[g] = ((long)(n * HH + yc) * WW + xc) * CHN + khalf;  // always in-bounds
        mask[g] = ok ? 1.0f : 0.0f;                               // masked at extraction
    }

    const float* b0 = f2t + base[0];
    const float* b1 = f2t + base[1];
    const float* b2 = f2t + base[2];

    v8f acc0 = {}, acc1 = {}, acc2 = {};
#pragma unroll 4
    for (int c0 = 0; c0 < CHN; c0 += 4) {
        v2f a  = *(const v2f*)(sF1 + c0 + khalf);   // f1 chunk, broadcast over rows
        v2f q0 = *(const v2f*)(b0 + c0);            // f2 chunk for position 16*0+nl
        v2f q1 = *(const v2f*)(b1 + c0);
        v2f q2 = *(const v2f*)(b2 + c0);
        acc0 = __builtin_amdgcn_wmma_f32_16x16x4_f32(false, a, false, q0, (short)0, acc0, false, false);
        acc1 = __builtin_amdgcn_wmma_f32_16x16x4_f32(false, a, false, q1, (short)0, acc1, false, false);
        acc2 = __builtin_amdgcn_wmma_f32_16x16x4_f32(false, a, false, q2, (short)0, acc2, false, false);
    }

    // D rows are identical (A broadcast): element 0 of each accumulator holds
    // the dot for position 16g + (lane&15). Lanes 0-15 publish to LDS.
    if (lane < 16) {
        sD[lane]      = acc0[0] * mask[0];
        sD[16 + lane] = acc1[0] * mask[1];
        if (lane < 4) sD[32 + lane] = acc2[0] * mask[2];
    }
    __syncthreads();

    // Bilinear 4-tap combine: tap (dy,dx) reads D[dy..dy+1][dx..dx+1].
    float cvv = -INFINITY;
    if (lane < 25) {
        const int dy = lane / 5, dx = lane % 5;
        const float w00 = (1.0f - wx) * (1.0f - wy);
        const float w10 = wx * (1.0f - wy);
        const float w01 = (1.0f - wx) * wy;
        const float w11 = wx * wy;
        cvv = w00 * sD[dy * 6 + dx]     + w10 * sD[dy * 6 + dx + 1]
            + w01 * sD[(dy + 1) * 6 + dx] + w11 * sD[(dy + 1) * 6 + dx + 1];
        out[(long)pix * 25 + lane] = cvv;
    }

    // WTA max + first-index argmax (matches jnp.argmax tie-break).
    float bv = cvv;
    int   bi = (lane < 25) ? lane : 64;
#pragma unroll
    for (int off = 16; off >= 1; off >>= 1) {
        float ov = __shfl_xor(bv, off, 32);
        int   oi = __shfl_xor(bi, off, 32);
        if (ov > bv || (ov == bv && oi < bi)) { bv = ov; bi = oi; }
    }
    if (lane == 0) {
        out[OUT_WTA + pix]      = bv;
        out[OUT_XY + pix]       = (float)(bi % 5);  // x offset index
        out[OUT_XY + NHW + pix] = (float)(bi / 5);  // y offset index
    }
}

// ---------------------------------------------------------------------------
// Launch: transpose f1 and f2 into d_ws (needs 2*N*C*H*W*4 = 28.8 MB), then
// one wave per pixel for the cost volume. All on `stream`, graph-capture safe.
// ---------------------------------------------------------------------------
extern "C" void kernel_launch(void* const* d_in, const int* in_sizes, int n_in,
                              void* d_out, int out_size, void* d_ws, size_t ws_size,
                              hipStream_t stream) {
    const float* flow = (const float*)d_in[0];
    const float* f1   = (const float*)d_in[1];
    const float* f2   = (const float*)d_in[2];
    float* f1t = (float*)d_ws;
    float* f2t = f1t + (long)NB * CHN * HH * WW;

    dim3 tb(32, 8);
    dim3 tg(32, NB * HH);   // 8 c-tiles * 4 w-tiles, one (n,h) plane per blockIdx.y
    xpose_kernel<<<tg, tb, 0, stream>>>(f1, f1t);
    xpose_kernel<<<tg, tb, 0, stream>>>(f2, f2t);

    cv_kernel<<<dim3(NHW), dim3(32), 0, stream>>>(flow, f1t, f2t, (float*)d_out);
}